// GaussianImage_Cholesky_87333864997144
// MI455X (gfx1250) — compile-verified
//
#include <hip/hip_runtime.h>
#include <math.h>

typedef __attribute__((ext_vector_type(2))) float v2f;
typedef __attribute__((ext_vector_type(8))) float v8f;

#define N_POINTS 512
#define IMG_H 512
#define IMG_W 512
#define N_CHUNKS (N_POINTS / 4)   // 128 K-chunks of 4 gaussians

// One block = 256 threads = 8 wave32. Each wave renders a strip of 16
// consecutive pixels in one image row against all 512 gaussians.
//
// WMMA mapping (V_WMMA_F32_16X16X4_F32, D = A(16x4) * B(4x16) + C):
//   A = alpha tile: M = pixel-in-strip (0..15), K = gaussian chunk (4 points)
//     f32 A layout (2 VGPRs): lanes 0-15 -> M=lane, K={0,1};
//                             lanes 16-31 -> M=lane-16, K={2,3}
//     => every lane computes exactly 2 alphas per chunk (full utilization)
//   B = features: K = gaussian, N = channel (cols 0..2 live, rest zero).
//     The per-lane B register pair for each chunk is precomputed into LDS
//     (sB) so the hot loop issues ONE unconditional ds_load_b64 for B.
//   C/D accumulator: VGPR r, lanes 0-15 -> M=r, col=lane;
//                    lanes 16-31 -> M=8+r, col=lane-16
__global__ __launch_bounds__(256) void gaussian_render_wmma(
    const float* __restrict__ xyz,
    const float* __restrict__ cholesky,
    const float* __restrict__ features,
    const float* __restrict__ opacity,
    float* __restrict__ out)
{
    // Per-gaussian params: {cx, cy, conic_a, conic_b, conic_c, opac, 0, 0}
    __shared__ __align__(16) float sParams[N_POINTS * 8];   // 16 KB
    // Pre-baked WMMA B operand per (chunk, lane): float2       32 KB
    __shared__ __align__(8) v2f sB[N_CHUNKS * 32];

    // ---- Stage 1: per-block precompute of gaussian params (one-time) ----
    for (int i = threadIdx.x; i < N_POINTS; i += 256) {
        float mx = tanhf(xyz[2 * i + 0]);
        float my = tanhf(xyz[2 * i + 1]);
        float l1 = cholesky[3 * i + 0] + 0.5f;
        float l2 = cholesky[3 * i + 1];
        float l3 = cholesky[3 * i + 2] + 0.5f;
        float s00 = l1 * l1;
        float s01 = l1 * l2;
        float s11 = l2 * l2 + l3 * l3;
        float det = s00 * s11 - s01 * s01;
        float inv = 1.0f / det;
        sParams[i * 8 + 0] = 0.5f * (float)IMG_W * (mx + 1.0f);  // cx
        sParams[i * 8 + 1] = 0.5f * (float)IMG_H * (my + 1.0f);  // cy
        sParams[i * 8 + 2] = s11 * inv;                          // conic_a
        sParams[i * 8 + 3] = -s01 * inv;                         // conic_b
        sParams[i * 8 + 4] = s00 * inv;                          // conic_c
        sParams[i * 8 + 5] = opacity[i];
        sParams[i * 8 + 6] = 0.0f;
        sParams[i * 8 + 7] = 0.0f;
    }
    // B operand image: chunk ch, lane ln -> gaussians kp, kp+1 at column col
    for (int idx = threadIdx.x; idx < N_CHUNKS * 32; idx += 256) {
        int ch  = idx >> 5;
        int ln  = idx & 31;
        int col = ln & 15;
        int kp  = ch * 4 + (ln >> 4) * 2;
        v2f b;
        b.x = (col < 3) ? features[3 * kp + col]       : 0.0f;
        b.y = (col < 3) ? features[3 * (kp + 1) + col] : 0.0f;
        sB[idx] = b;
    }
    __syncthreads();

    const int lane = threadIdx.x & 31;
    const int wave = threadIdx.x >> 5;
    const int strip = blockIdx.x * 8 + wave;   // 16-pixel strip index
    const int pixBase = strip << 4;
    const int y  = pixBase >> 9;               // / IMG_W (512)
    const int x0 = pixBase & (IMG_W - 1);
    const int m  = lane & 15;                  // pixel-in-strip (A row M)
    const int kh = lane >> 4;                  // 0 -> K pair {0,1}, 1 -> {2,3}

    const float px = (float)(x0 + m) + 0.5f;
    const float py = (float)y + 0.5f;

    const float LOG2E  = 1.4426950408889634f;
    const float CLAMP  = 0.999f;
    const float THRESH = 1.0f / 255.0f;

    v8f acc = {0.f, 0.f, 0.f, 0.f, 0.f, 0.f, 0.f, 0.f};

    // ---- Stage 2: hot loop, K=512 in chunks of 4 ----
    #pragma unroll 4
    for (int k0 = 0; k0 < N_POINTS; k0 += 4) {
        const int kp = k0 + kh * 2;            // this lane's first gaussian
        const float* p0 = &sParams[kp * 8];    // ds_load_b128 (+b64)
        const float* p1 = &sParams[kp * 8 + 8];

        // alpha for gaussian kp
        float dx0 = p0[0] - px;
        float dy0 = p0[1] - py;
        float sg0 = 0.5f * (p0[2] * dx0 * dx0 + p0[4] * dy0 * dy0)
                  + p0[3] * dx0 * dy0;
        float al0 = p0[5] * __builtin_amdgcn_exp2f(-sg0 * LOG2E);
        al0 = fminf(al0, CLAMP);
        al0 = (sg0 < 0.0f) ? 0.0f : al0;
        al0 = (al0 < THRESH) ? 0.0f : al0;

        // alpha for gaussian kp+1
        float dx1 = p1[0] - px;
        float dy1 = p1[1] - py;
        float sg1 = 0.5f * (p1[2] * dx1 * dx1 + p1[4] * dy1 * dy1)
                  + p1[3] * dx1 * dy1;
        float al1 = p1[5] * __builtin_amdgcn_exp2f(-sg1 * LOG2E);
        al1 = fminf(al1, CLAMP);
        al1 = (sg1 < 0.0f) ? 0.0f : al1;
        al1 = (al1 < THRESH) ? 0.0f : al1;

        v2f A;
        A.x = al0;
        A.y = al1;

        // One unconditional ds_load_b64: pre-baked B operand for this chunk
        v2f B = sB[(k0 >> 2) * 32 + lane];

        // 8 args: (neg_a, A, neg_b, B, c_mod, C, reuse_a, reuse_b)
        acc = __builtin_amdgcn_wmma_f32_16x16x4_f32(
            false, A, false, B, (short)0, acc, false, false);
    }

    // ---- Write-out: lanes 0-2 -> channel=lane, pixels x0+0..7 (r=M);
    //                 lanes 16-18 -> channel=lane-16, pixels x0+8..15. ----
    if (m < 3) {
        float* o = out + (size_t)m * (IMG_H * IMG_W)
                       + (size_t)y * IMG_W + x0 + kh * 8;
        #pragma unroll
        for (int r = 0; r < 8; ++r) {
            float v = acc[r];
            v = fminf(fmaxf(v, 0.0f), 1.0f);
            o[r] = v;
        }
    }
}

extern "C" void kernel_launch(void* const* d_in, const int* in_sizes, int n_in,
                              void* d_out, int out_size, void* d_ws, size_t ws_size,
                              hipStream_t stream) {
    (void)in_sizes; (void)n_in; (void)out_size; (void)d_ws; (void)ws_size;
    const float* xyz      = (const float*)d_in[0];
    const float* cholesky = (const float*)d_in[1];
    const float* features = (const float*)d_in[2];
    const float* opacity  = (const float*)d_in[3];
    // d_in[4]/d_in[5] are H, W scalars; fixed at 512 per the reference constants.
    float* out = (float*)d_out;

    const int strips = (IMG_H * IMG_W) / 16;   // 16384
    const int blocks = strips / 8;             // 2048 blocks x 256 threads
    gaussian_render_wmma<<<blocks, 256, 0, stream>>>(
        xyz, cholesky, features, opacity, out);
}